// Attention_18528488915489
// MI455X (gfx1250) — compile-verified
//
#include <hip/hip_runtime.h>
#include <math.h>

// Problem constants (match reference)
#define T_N 65536
#define B_N 512
#define D_N 64      // D1 == D2 == 64
#define A_N 256
#define F_P_BASE 64     // W1 rows: [0,64)=x, [64,4160)=p, [4160,4224)=q
#define F_Q_BASE 4160

typedef __attribute__((ext_vector_type(2))) float v2f;
typedef __attribute__((ext_vector_type(8))) float v8f;

static __device__ __forceinline__ v8f wmma4(v2f a, v2f b, v8f c) {
  // V_WMMA_F32_16X16X4_F32 : D = A(16x4) * B(4x16) + C(16x16), all fp32
  return __builtin_amdgcn_wmma_f32_16x16x4_f32(
      /*neg_a=*/false, a, /*neg_b=*/false, b,
      /*c_mod=*/(short)0, c, /*reuse_a=*/false, /*reuse_b=*/false);
}

// ---------------- sort-by-batch (counting sort) ----------------
__global__ void k_zero_hist(int* hist) {
  int i = threadIdx.x;
  if (i < B_N) hist[i] = 0;
}

__global__ void k_hist(const int* __restrict__ g, int* __restrict__ hist) {
  int t = blockIdx.x * blockDim.x + threadIdx.x;
  if (t < T_N) atomicAdd(&hist[g[t]], 1);
}

__global__ void k_scan(const int* __restrict__ hist, int* __restrict__ base,
                       int* __restrict__ cursor) {
  if (threadIdx.x == 0) {
    int run = 0;
    for (int b = 0; b < B_N; ++b) {
      base[b] = run;
      cursor[b] = run;
      run += hist[b];
    }
  }
}

__global__ void k_scatter(const int* __restrict__ g, int* __restrict__ cursor,
                          int* __restrict__ order) {
  int t = blockIdx.x * blockDim.x + threadIdx.x;
  if (t < T_N) {
    int pos = atomicAdd(&cursor[g[t]], 1);
    order[pos] = t;
  }
}

// ---------------- stage 1: M'[b,i,:] = W1_x[i,:] + query[b,:]·W1_p[i,:,:]
//                  c[b,:]   = b1 + query[b,:]·W1_q
// grid = (65, 32): blockIdx.x = i-slot (64 == c-slot), blockIdx.y = b-tile.
__global__ void k_build_m(const float* __restrict__ query,
                          const float* __restrict__ W1,
                          const float* __restrict__ b1,
                          float* __restrict__ Mws, float* __restrict__ cws) {
  const int i = blockIdx.x;
  const int btile = blockIdx.y;
  const int w = threadIdx.x >> 5;
  const int lane = threadIdx.x & 31;
  const int hi = lane >> 4;
  const int ln = lane & 15;
  const int col0 = w * 32 + ln;
  const int col1 = col0 + 16;
  const int bm = btile * 16 + ln;  // query row for A-matrix row (lane&15)

  const float* Wsrc =
      W1 + (size_t)((i < 64) ? (F_P_BASE + i * 64) : F_Q_BASE) * A_N;

  v8f c0 = {}, c1 = {};
#pragma unroll
  for (int s = 0; s < 16; ++s) {
    const int kk = 4 * s + 2 * hi;
    v2f a = *(const v2f*)(query + (size_t)bm * D_N + kk);
    v2f bv0, bv1;
    bv0[0] = Wsrc[(size_t)kk * A_N + col0];
    bv0[1] = Wsrc[(size_t)(kk + 1) * A_N + col0];
    bv1[0] = Wsrc[(size_t)kk * A_N + col1];
    bv1[1] = Wsrc[(size_t)(kk + 1) * A_N + col1];
    c0 = wmma4(a, bv0, c0);
    c1 = wmma4(a, bv1, c1);
  }

  float bias0, bias1;
  if (i < 64) {  // fold W1_x row i into M'
    bias0 = W1[(size_t)i * A_N + col0];
    bias1 = W1[(size_t)i * A_N + col1];
  } else {  // fold b1 into c
    bias0 = b1[col0];
    bias1 = b1[col1];
  }

#pragma unroll
  for (int rr = 0; rr < 8; ++rr) {
    const int bm2 = btile * 16 + rr + 8 * hi;  // batch row for C row
    if (i < 64) {
      Mws[((size_t)bm2 * 64 + i) * A_N + col0] = c0[rr] + bias0;
      Mws[((size_t)bm2 * 64 + i) * A_N + col1] = c1[rr] + bias1;
    } else {
      cws[(size_t)bm2 * A_N + col0] = c0[rr] + bias0;
      cws[(size_t)bm2 * A_N + col1] = c1[rr] + bias1;
    }
  }
}

// ---------------- stage 2: grouped GEMM  z[t,:] = x[t,:]·M'[b] + c[b]
// one block per batch b; 8 waves x 2 N-tiles; B-panel register-resident.
__global__ void k_gemm(const float* __restrict__ x,
                       const float* __restrict__ Mws,
                       const float* __restrict__ cws,
                       const int* __restrict__ hist,
                       const int* __restrict__ base,
                       const int* __restrict__ order,
                       float* __restrict__ zws, float* __restrict__ psum,
                       float* __restrict__ psq) {
  const int b = blockIdx.x;
  const int cnt = hist[b];
  const int bs = base[b];
  const int w = threadIdx.x >> 5;
  const int lane = threadIdx.x & 31;
  const int hi = lane >> 4;
  const int ln = lane & 15;
  const int col0 = w * 32 + ln;
  const int col1 = col0 + 16;

  if (cnt == 0) {  // keep stats arrays defined
    psum[(size_t)b * A_N + threadIdx.x] = 0.f;
    psq[(size_t)b * A_N + threadIdx.x] = 0.f;
    return;
  }

  // Load full K=64 B-panel for both N-tiles into registers (L2-hot).
  float bf0[16][2], bf1[16][2];
#pragma unroll
  for (int s = 0; s < 16; ++s) {
    const int kk = 4 * s + 2 * hi;
    bf0[s][0] = Mws[((size_t)b * 64 + kk) * A_N + col0];
    bf0[s][1] = Mws[((size_t)b * 64 + kk + 1) * A_N + col0];
    bf1[s][0] = Mws[((size_t)b * 64 + kk) * A_N + col1];
    bf1[s][1] = Mws[((size_t)b * 64 + kk + 1) * A_N + col1];
  }
  const float cb0 = cws[(size_t)b * A_N + col0];
  const float cb1 = cws[(size_t)b * A_N + col1];

  float s0 = 0.f, q0 = 0.f, s1 = 0.f, q1 = 0.f;
  const int mtiles = (cnt + 15) >> 4;
  const int rlast = bs + cnt - 1;

  for (int mt = 0; mt < mtiles; ++mt) {
    int ridx = bs + mt * 16 + ln;
    const int tA = order[ridx < rlast ? ridx : rlast];  // clamp tail rows
    v8f acc0 = {}, acc1 = {};
#pragma unroll
    for (int s = 0; s < 16; ++s) {
      const int kk = 4 * s + 2 * hi;
      v2f av = *(const v2f*)(x + (size_t)tA * D_N + kk);
      v2f bv0, bv1;
      bv0[0] = bf0[s][0]; bv0[1] = bf0[s][1];
      bv1[0] = bf1[s][0]; bv1[1] = bf1[s][1];
      acc0 = wmma4(av, bv0, acc0);
      acc1 = wmma4(av, bv1, acc1);
    }
#pragma unroll
    for (int rr = 0; rr < 8; ++rr) {
      const int gr = mt * 16 + rr + 8 * hi;
      if (gr < cnt) {
        const int tS = order[bs + gr];
        const float v0 = acc0[rr] + cb0;
        const float v1 = acc1[rr] + cb1;
        zws[(size_t)tS * A_N + col0] = v0;
        zws[(size_t)tS * A_N + col1] = v1;
        s0 += v0; q0 += v0 * v0;
        s1 += v1; q1 += v1 * v1;
      }
    }
  }

  // Combine the two half-wave row groups (same column) and emit per-block
  // partials; each wave owns disjoint columns -> no atomics, deterministic.
  const float t0 = s0 + __shfl_xor(s0, 16);
  const float u0 = q0 + __shfl_xor(q0, 16);
  const float t1 = s1 + __shfl_xor(s1, 16);
  const float u1 = q1 + __shfl_xor(q1, 16);
  if (hi == 0) {
    psum[(size_t)b * A_N + col0] = t0;
    psq[(size_t)b * A_N + col0] = u0;
    psum[(size_t)b * A_N + col1] = t1;
    psq[(size_t)b * A_N + col1] = u1;
  }
}

// ---------------- stage 2b: column mean / rstd (unbiased var, EPS=1e-9)
__global__ void k_stats(const float* __restrict__ psum,
                        const float* __restrict__ psq,
                        float* __restrict__ meanv, float* __restrict__ rstdv) {
  const int a = threadIdx.x;
  float s = 0.f, q = 0.f;
  for (int b = 0; b < B_N; ++b) {
    s += psum[(size_t)b * A_N + a];
    q += psq[(size_t)b * A_N + a];
  }
  const float mean = s / (float)T_N;
  float var = (q - (float)T_N * mean * mean) / (float)(T_N - 1);
  var = fmaxf(var, 0.f);
  meanv[a] = mean;
  rstdv[a] = rsqrtf(var + 1e-9f);
}

// ---------------- stage 3: Dice gate + z@W2 + b2, one wave per t
__global__ void k_out(const float* __restrict__ zws,
                      const float* __restrict__ meanv,
                      const float* __restrict__ rstdv,
                      const float* __restrict__ W2,
                      const float* __restrict__ alphap,
                      const float* __restrict__ b2p, float* __restrict__ out) {
  const int w = threadIdx.x >> 5;
  const int lane = threadIdx.x & 31;
  const int t = blockIdx.x * 8 + w;
  const float alpha = alphap[0];
  float acc = 0.f;
#pragma unroll
  for (int j = 0; j < 8; ++j) {
    const int a = lane + 32 * j;
    const float zv = zws[(size_t)t * A_N + a];
    const float nrm = (zv - meanv[a]) * rstdv[a];
    const float p = 1.f / (1.f + expf(-nrm));
    const float val = zv * (p + alpha * (1.f - p));
    acc += val * W2[a];
  }
  acc += __shfl_xor(acc, 16);
  acc += __shfl_xor(acc, 8);
  acc += __shfl_xor(acc, 4);
  acc += __shfl_xor(acc, 2);
  acc += __shfl_xor(acc, 1);
  if (lane == 0) out[t] = acc + b2p[0];
}

// ---------------- host launcher ----------------
extern "C" void kernel_launch(void* const* d_in, const int* in_sizes, int n_in,
                              void* d_out, int out_size, void* d_ws,
                              size_t ws_size, hipStream_t stream) {
  (void)in_sizes; (void)n_in; (void)out_size; (void)ws_size;
  const float* x     = (const float*)d_in[0];
  const float* query = (const float*)d_in[1];
  const int*   gidx  = (const int*)d_in[2];
  const float* W1    = (const float*)d_in[3];
  const float* b1    = (const float*)d_in[4];
  const float* alpha = (const float*)d_in[5];
  const float* W2    = (const float*)d_in[6];
  const float* b2    = (const float*)d_in[7];
  float* out = (float*)d_out;

  // Workspace carve-up (all 256B aligned)
  char* ws = (char*)d_ws;
  size_t off = 0;
  auto take = [&](size_t bytes) {
    void* p = ws + off;
    off += (bytes + 255) & ~(size_t)255;
    return p;
  };
  int*   hist   = (int*)take(B_N * sizeof(int));
  int*   basep  = (int*)take(B_N * sizeof(int));
  int*   cursor = (int*)take(B_N * sizeof(int));
  int*   order  = (int*)take((size_t)T_N * sizeof(int));
  float* Mws    = (float*)take((size_t)B_N * D_N * A_N * sizeof(float)); // 33.5 MB
  float* cws    = (float*)take((size_t)B_N * A_N * sizeof(float));
  float* zws    = (float*)take((size_t)T_N * A_N * sizeof(float));       // 67 MB
  float* psum   = (float*)take((size_t)B_N * A_N * sizeof(float));
  float* psq    = (float*)take((size_t)B_N * A_N * sizeof(float));
  float* meanv  = (float*)take(A_N * sizeof(float));
  float* rstdv  = (float*)take(A_N * sizeof(float));

  // 1) counting sort of timesteps by batch id
  hipLaunchKernelGGL(k_zero_hist, dim3(1), dim3(512), 0, stream, hist);
  hipLaunchKernelGGL(k_hist, dim3(T_N / 256), dim3(256), 0, stream, gidx, hist);
  hipLaunchKernelGGL(k_scan, dim3(1), dim3(1), 0, stream, hist, basep, cursor);
  hipLaunchKernelGGL(k_scatter, dim3(T_N / 256), dim3(256), 0, stream, gidx,
                     cursor, order);
  // 2) per-batch effective weights M' and bias c
  hipLaunchKernelGGL(k_build_m, dim3(65, 32), dim3(256), 0, stream, query, W1,
                     b1, Mws, cws);
  // 3) grouped WMMA GEMM -> z + per-block column partial sums
  hipLaunchKernelGGL(k_gemm, dim3(B_N), dim3(256), 0, stream, x, Mws, cws,
                     hist, basep, order, zws, psum, psq);
  // 4) batch statistics
  hipLaunchKernelGGL(k_stats, dim3(1), dim3(A_N), 0, stream, psum, psq, meanv,
                     rstdv);
  // 5) Dice + final projection
  hipLaunchKernelGGL(k_out, dim3(T_N / 8), dim3(256), 0, stream, zws, meanv,
                     rstdv, W2, alpha, b2, out);
}